// MultiHeadAttention_81398220194213
// MI455X (gfx1250) — compile-verified
//
#include <hip/hip_runtime.h>
#include <math.h>
#include <stdint.h>

// Problem constants (reference: B=8, S=512, D=1024, H=16, DK=64, N_ASP=4)
#define BB_ 8
#define SS_ 512
#define DD_ 1024
#define HH_ 16
#define DKK_ 64
#define NEGV -1000000000.0f
#define APAD 68   // padded LDS row stride (floats) -> conflict-free A-frag reads

typedef __attribute__((ext_vector_type(2))) float v2f;
typedef __attribute__((ext_vector_type(8))) float v8f;

// Raw LDS byte offset of a __shared__ object: low 32 bits of the generic
// address are the LDS offset (ISA 10.2).
__device__ __forceinline__ uint32_t lds_off(const void* p) {
  return (uint32_t)(uintptr_t)p;
}

// Async-copy 16 bytes from global to LDS (GLOBAL_LOAD_ASYNC_TO_LDS_B128,
// tracked by ASYNCcnt), then drain this wave's async queue.
__device__ __forceinline__ void async_copy_b128(uint32_t lds_byte_addr,
                                                const float* gsrc) {
  asm volatile("global_load_async_to_lds_b128 %0, %1, off"
               :: "v"(lds_byte_addr), "v"(gsrc)
               : "memory");
}
__device__ __forceinline__ void async_wait0() {
  asm volatile("s_wait_asynccnt 0x0" ::: "memory");
}

// ---------------------------------------------------------------------------
// Kernel 1: projection GEMM  out[b,h,s,dk] = X[row,:]@W[:,col] + bias[col]
// M = B*S = 4096, N = D = 1024, K = D = 1024.
// Block = 8 waves; all waves share one 16-row A strip. Per 64-wide K chunk,
// the block async-stages the 16x64 A tile into LDS (256 threads x 16B), then
// each wave runs 16 V_WMMA_F32_16X16X4_F32 steps reading A from LDS and its
// own W columns from global.
// A-frag layout (ISA 7.12.2, 32-bit A 16x4): lane<16 holds K=0,1; lane>=16 K=2,3.
// C/D layout: VGPR r -> rows r (lanes 0-15) and r+8 (lanes 16-31), N = lane&15.
// ---------------------------------------------------------------------------
__global__ __launch_bounds__(256) void proj_kernel(const float* __restrict__ X,
                                                   const float* __restrict__ W,
                                                   const float* __restrict__ bias,
                                                   float* __restrict__ out) {
  __shared__ float atile[16][APAD];   // 16 rows x 64 K (padded)

  const int tid  = threadIdx.x;
  const int wave = tid >> 5;
  const int lane = tid & 31;
  const int row0 = blockIdx.x * 16;                 // 256 row tiles
  const int col0 = (blockIdx.y * 8 + wave) * 16;    // 8 col tiles per block

  const int m  = lane & 15;
  const int kk = (lane >> 4) * 2;  // 0 or 2
  const int n  = lane & 15;

  // async staging assignment: thread -> (row, 4-float chunk)
  const int ar  = tid >> 4;          // 0..15
  const int ac4 = (tid & 15) * 4;    // 0,4,...,60
  const float* __restrict__ astage = X + (size_t)(row0 + ar) * DD_ + ac4;
  const uint32_t albase = lds_off(&atile[0][0]) + (uint32_t)(ar * APAD + ac4) * 4u;

  const float* __restrict__ bcol = W + col0 + n;

  v8f c = {0.f, 0.f, 0.f, 0.f, 0.f, 0.f, 0.f, 0.f};

  for (int kc = 0; kc < DD_; kc += 64) {
    // stage A tile for this K chunk (4 KB total, 16 B per thread)
    async_copy_b128(albase, astage + kc);
    __builtin_prefetch(astage + kc + 64, 0, 0);   // global_prefetch_b8
    async_wait0();
    __syncthreads();

#pragma unroll
    for (int k0 = 0; k0 < 64; k0 += 4) {
      v2f a, b;
      a.x = atile[m][k0 + kk];
      a.y = atile[m][k0 + kk + 1];
      b.x = bcol[(size_t)(kc + k0 + kk) * DD_];
      b.y = bcol[(size_t)(kc + k0 + kk + 1) * DD_];
      c = __builtin_amdgcn_wmma_f32_16x16x4_f32(false, a, false, b, (short)0, c,
                                                false, false);
    }
    __syncthreads();   // protect atile before next chunk overwrites it
  }

  const int col = col0 + n;
  const float bv = bias[col];
  const int h  = col >> 6;   // / DK
  const int dk = col & 63;   // % DK
#pragma unroll
  for (int r = 0; r < 8; ++r) {
    const int mrow = (lane < 16) ? r : (r + 8);
    const int grow = row0 + mrow;       // global row in [0, B*S)
    const int b_   = grow >> 9;         // / S
    const int s_   = grow & 511;        // % S
    out[(((size_t)b_ * HH_ + h) * SS_ + s_) * DKK_ + dk] = c[r] + bv;
  }
}

// ---------------------------------------------------------------------------
// Kernel 2: asp_vec[b,e] = aspect[b,:]@Wd[:,e] + bd[e]   (tiny)
// ---------------------------------------------------------------------------
__global__ void aspvec_kernel(const float* __restrict__ aspect,
                              const float* __restrict__ Wd,
                              const float* __restrict__ bd,
                              float* __restrict__ asp_vec) {
  const int b = blockIdx.x;
  const int e = threadIdx.x;  // 64 threads
  float acc = 0.f;
  for (int d = 0; d < DD_; ++d)
    acc += aspect[b * DD_ + d] * Wd[d * DKK_ + e];
  asp_vec[b * DKK_ + e] = acc + bd[e];
}

// ---------------------------------------------------------------------------
// Kernel 3: aw[b,h,e] = sum_d asp_vec[b,d] * weight_m[h,d,e]   (tiny)
// ---------------------------------------------------------------------------
__global__ void aw_kernel(const float* __restrict__ asp_vec,
                          const float* __restrict__ weight_m,
                          float* __restrict__ aw) {
  const int bh = blockIdx.x;  // B*H
  const int e  = threadIdx.x; // 64
  const int b = bh / HH_;
  const int h = bh % HH_;
  float acc = 0.f;
#pragma unroll 8
  for (int d = 0; d < DKK_; ++d)
    acc += asp_vec[b * DKK_ + d] * weight_m[(h * DKK_ + d) * DKK_ + e];
  aw[bh * DKK_ + e] = acc;
}

// ---------------------------------------------------------------------------
// Kernel 4: v[b,h,t] = tanh(aw[b,h,:]·k[b,h,t,:] + bias_m)
// (asps is s-independent; v doubles as the reference's `avg`.)
// ---------------------------------------------------------------------------
__global__ __launch_bounds__(256) void v_kernel(const float* __restrict__ aw,
                                                const float* __restrict__ k_ws,
                                                const float* __restrict__ bias_m,
                                                float* __restrict__ v) {
  const int bh = blockIdx.x;  // B*H
  __shared__ float saw[DKK_];
  if (threadIdx.x < DKK_) saw[threadIdx.x] = aw[bh * DKK_ + threadIdx.x];
  __syncthreads();
  const float bm = bias_m[0];
  for (int t = threadIdx.x; t < SS_; t += blockDim.x) {
    const float* kr = k_ws + ((size_t)bh * SS_ + t) * DKK_;
    float acc = 0.f;
#pragma unroll
    for (int e = 0; e < DKK_; ++e) acc += saw[e] * kr[e];
    v[bh * SS_ + t] = tanhf(acc + bm);
  }
}

// ---------------------------------------------------------------------------
// Kernel 5: a_sc fill (128 MB, bandwidth bound).
// Base value v[t]; scanning j=3..0, first match t==id_j -> v[s] (col set wins
// over row set within an iteration), else s==id_j -> v[t].
// ---------------------------------------------------------------------------
__global__ __launch_bounds__(256) void asc_kernel(const float* __restrict__ v,
                                                  const int* __restrict__ aspect_ids,
                                                  float* __restrict__ out) {
  const long long idx = (long long)blockIdx.x * 256 + threadIdx.x;
  const int t = (int)(idx & (SS_ - 1));
  const long long r = idx >> 9;
  const int s = (int)(r & (SS_ - 1));
  const long long bh = r >> 9;
  const int b = (int)(bh >> 4);  // / H

  const int id0 = aspect_ids[b * 4 + 0];
  const int id1 = aspect_ids[b * 4 + 1];
  const int id2 = aspect_ids[b * 4 + 2];
  const int id3 = aspect_ids[b * 4 + 3];

  const float* vb = v + bh * SS_;
  const float vt = vb[t];
  const float vs = vb[s];

  float val;
  if      (t == id3) val = vs;
  else if (s == id3) val = vt;
  else if (t == id2) val = vs;
  else if (s == id2) val = vt;
  else if (t == id1) val = vs;
  else if (s == id1) val = vt;
  else if (t == id0) val = vs;
  else if (s == id0) val = vt;
  else               val = vt;

  out[idx] = val;
}

// ---------------------------------------------------------------------------
// Kernel 6: fused scores + mask + softmax -> attn.
// One block per (b, h, 16-row s-tile). The shared 16x64 q tile is async-staged
// into LDS once; 8 waves each compute four 16x16 score tiles (cols 64w..64w+63)
// with V_WMMA_F32_16X16X4_F32 (DK=64 -> 16 k-steps), A-frags from LDS.
// Masked, scaled scores staged in 32KB LDS, then cooperative row softmax
// (16 threads per row) writes attn exactly once.
// ---------------------------------------------------------------------------
__global__ __launch_bounds__(256) void attn_kernel(const float* __restrict__ q_ws,
                                                   const float* __restrict__ k_ws,
                                                   const int* __restrict__ mask,
                                                   float* __restrict__ attn_out) {
  __shared__ float sc[16][SS_];       // 32 KB
  __shared__ float red[16][16];       // 1 KB
  __shared__ float qtile[16][APAD];   // ~4.25 KB

  const int b = blockIdx.z;
  const int h = blockIdx.y;
  const int s0 = blockIdx.x * 16;

  const int tid  = threadIdx.x;
  const int wave = tid >> 5;
  const int lane = tid & 31;
  const int m  = lane & 15;
  const int kk = (lane >> 4) * 2;
  const int n  = lane & 15;

  // Async-stage the 16x64 q tile shared by all 8 waves.
  {
    const int ar  = tid >> 4;          // 0..15
    const int ac4 = (tid & 15) * 4;    // 0..60
    const float* g =
        q_ws + (((size_t)b * HH_ + h) * SS_ + s0 + ar) * DKK_ + ac4;
    const uint32_t la =
        lds_off(&qtile[0][0]) + (uint32_t)(ar * APAD + ac4) * 4u;
    async_copy_b128(la, g);
    async_wait0();
  }
  __syncthreads();

  const float* __restrict__ kbase = k_ws + ((size_t)b * HH_ + h) * SS_ * DKK_;

  v8f acc0 = {0.f,0.f,0.f,0.f,0.f,0.f,0.f,0.f};
  v8f acc1 = acc0, acc2 = acc0, acc3 = acc0;

#pragma unroll 4
  for (int k0 = 0; k0 < DKK_; k0 += 4) {
    v2f a;
    a.x = qtile[m][k0 + kk];
    a.y = qtile[m][k0 + kk + 1];
    {
      const float* kr = kbase + (size_t)((wave * 4 + 0) * 16 + n) * DKK_;
      v2f bf; bf.x = kr[k0 + kk]; bf.y = kr[k0 + kk + 1];
      acc0 = __builtin_amdgcn_wmma_f32_16x16x4_f32(false, a, false, bf, (short)0, acc0, false, false);
    }
    {
      const float* kr = kbase + (size_t)((wave * 4 + 1) * 16 + n) * DKK_;
      v2f bf; bf.x = kr[k0 + kk]; bf.y = kr[k0 + kk + 1];
      acc1 = __builtin_amdgcn_wmma_f32_16x16x4_f32(false, a, false, bf, (short)0, acc1, false, false);
    }
    {
      const float* kr = kbase + (size_t)((wave * 4 + 2) * 16 + n) * DKK_;
      v2f bf; bf.x = kr[k0 + kk]; bf.y = kr[k0 + kk + 1];
      acc2 = __builtin_amdgcn_wmma_f32_16x16x4_f32(false, a, false, bf, (short)0, acc2, false, false);
    }
    {
      const float* kr = kbase + (size_t)((wave * 4 + 3) * 16 + n) * DKK_;
      v2f bf; bf.x = kr[k0 + kk]; bf.y = kr[k0 + kk + 1];
      acc3 = __builtin_amdgcn_wmma_f32_16x16x4_f32(false, a, false, bf, (short)0, acc3, false, false);
    }
  }

  // Epilogue: scale 1/sqrt(64)=0.125, apply mask, stage to LDS.
  const float scale = 0.125f;
#pragma unroll
  for (int tt = 0; tt < 4; ++tt) {
    const v8f acc = (tt == 0) ? acc0 : (tt == 1) ? acc1 : (tt == 2) ? acc2 : acc3;
    const int col = (wave * 4 + tt) * 16 + n;
#pragma unroll
    for (int r = 0; r < 8; ++r) {
      const int mrow = (lane < 16) ? r : (r + 8);
      const int srow = s0 + mrow;
      const int mk = mask[((size_t)b * SS_ + srow) * SS_ + col];
      sc[mrow][col] = (mk == 0) ? NEGV : acc[r] * scale;
    }
  }
  __syncthreads();

  // Row softmax: 16 threads per row.
  const int row = tid >> 4;
  const int i   = tid & 15;

  float mx = -1e30f;
  for (int c = i; c < SS_; c += 16) mx = fmaxf(mx, sc[row][c]);
  red[row][i] = mx;
  __syncthreads();
  mx = -1e30f;
#pragma unroll
  for (int j = 0; j < 16; ++j) mx = fmaxf(mx, red[row][j]);
  __syncthreads();  // everyone done reading red before it is reused

  float sum = 0.f;
  for (int c = i; c < SS_; c += 16) {
    const float e = __expf(sc[row][c] - mx);
    sc[row][c] = e;
    sum += e;
  }
  red[row][i] = sum;
  __syncthreads();
  float tot = 0.f;
#pragma unroll
  for (int j = 0; j < 16; ++j) tot += red[row][j];
  const float inv = 1.0f / tot;

  float* __restrict__ orow =
      attn_out + ((size_t)(b * HH_ + h) * SS_ + s0 + row) * SS_;
  for (int c = i; c < SS_; c += 16) orow[c] = sc[row][c] * inv;
}

// ---------------------------------------------------------------------------
// Launch
// ---------------------------------------------------------------------------
extern "C" void kernel_launch(void* const* d_in, const int* in_sizes, int n_in,
                              void* d_out, int out_size, void* d_ws, size_t ws_size,
                              hipStream_t stream) {
  const float* query      = (const float*)d_in[0];
  const float* key_in     = (const float*)d_in[1];
  const int*   mask       = (const int*)  d_in[2];
  const float* aspect     = (const float*)d_in[3];
  const int*   aspect_ids = (const int*)  d_in[4];
  const float* Wq         = (const float*)d_in[5];
  const float* bq         = (const float*)d_in[6];
  const float* Wk         = (const float*)d_in[7];
  const float* bk         = (const float*)d_in[8];
  const float* Wd         = (const float*)d_in[9];
  const float* bd         = (const float*)d_in[10];
  const float* weight_m   = (const float*)d_in[11];
  const float* bias_m     = (const float*)d_in[12];

  float* a_sc = (float*)d_out;                                   // 8*16*512*512
  float* attn = (float*)d_out + (size_t)BB_ * HH_ * SS_ * SS_;   // second output

  float* q_ws    = (float*)d_ws;                                 // B*H*S*DK
  float* k_ws    = q_ws + (size_t)BB_ * HH_ * SS_ * DKK_;
  float* asp_vec = k_ws + (size_t)BB_ * HH_ * SS_ * DKK_;        // B*DK
  float* aw      = asp_vec + BB_ * DKK_;                         // B*H*DK
  float* vv      = aw + BB_ * HH_ * DKK_;                        // B*H*S

  // Projections: q and k
  proj_kernel<<<dim3(256, 8), 256, 0, stream>>>(query, Wq, bq, q_ws);
  proj_kernel<<<dim3(256, 8), 256, 0, stream>>>(key_in, Wk, bk, k_ws);

  // Aspect path (tiny)
  aspvec_kernel<<<BB_, DKK_, 0, stream>>>(aspect, Wd, bd, asp_vec);
  aw_kernel<<<BB_ * HH_, DKK_, 0, stream>>>(asp_vec, weight_m, aw);
  v_kernel<<<BB_ * HH_, 256, 0, stream>>>(aw, k_ws, bias_m, vv);

  // a_sc fill (bandwidth bound)
  const long long total = (long long)BB_ * HH_ * SS_ * SS_;
  asc_kernel<<<(unsigned)(total / 256), 256, 0, stream>>>(vv, aspect_ids, a_sc);

  // Fused scores + softmax
  attn_kernel<<<dim3(SS_ / 16, HH_, BB_), 256, 0, stream>>>(q_ws, k_ws, mask, attn);
}